// PM_Encoder_53970559041955
// MI455X (gfx1250) — compile-verified
//
#include <hip/hip_runtime.h>
#include <hip/hip_bf16.h>
#include <math.h>

typedef float v2f __attribute__((ext_vector_type(2)));
typedef float v8f __attribute__((ext_vector_type(8)));

// ---------------------------------------------------------------------------
// Degree / normalization kernels
// ---------------------------------------------------------------------------
__global__ void k_init_deg(float* __restrict__ deg, int n) {
  int i = blockIdx.x * blockDim.x + threadIdx.x;
  if (i < n) deg[i] = 1.0f;  // self-loop contributes weight 1 to every node
}

__global__ void k_deg_edges(const int* __restrict__ dst, const float* __restrict__ ew,
                            float* __restrict__ deg, int e) {
  int i = blockIdx.x * blockDim.x + threadIdx.x;
  if (i < e) atomicAdd(deg + dst[i], ew[i]);
}

__global__ void k_dinv(float* __restrict__ deg, int n) {
  int i = blockIdx.x * blockDim.x + threadIdx.x;
  if (i < n) {
    float d = deg[i];
    deg[i] = d > 0.0f ? rsqrtf(d) : 0.0f;  // in place: deg -> dinv
  }
}

__global__ void k_edge_norm(const int* __restrict__ src, const int* __restrict__ dst,
                            const float* __restrict__ ew, const float* __restrict__ dinv,
                            float* __restrict__ norm, int e) {
  int i = blockIdx.x * blockDim.x + threadIdx.x;
  if (i < e) norm[i] = dinv[src[i]] * ew[i] * dinv[dst[i]];
}

// ---------------------------------------------------------------------------
// f32 WMMA GEMM: C[M x Nd] = A[M x K] @ B[K x Nd].  M%16==0, K%4==0, Nd%16==0.
// One wave computes one 16x16 tile via V_WMMA_F32_16X16X4_F32.
// block = 128 threads (4 waves), grid = (M/16, Nd/64)
// ---------------------------------------------------------------------------
__global__ __launch_bounds__(128) void k_gemm_f32_wmma(
    const float* __restrict__ A, const float* __restrict__ B, float* __restrict__ C,
    int M, int K, int Nd) {
  const int wave  = threadIdx.x >> 5;
  const int lane  = threadIdx.x & 31;
  const int tileM = blockIdx.x;
  const int tileN = blockIdx.y * 4 + wave;
  if (tileN * 16 >= Nd) return;  // wave-uniform

  const int m     = tileM * 16 + (lane & 15);
  const int n     = tileN * 16 + (lane & 15);
  const int khalf = (lane >> 4) * 2;  // lanes 0-15 -> K {0,1}; lanes 16-31 -> K {2,3}

  v8f c = {};
  for (int kk = 0; kk < K; kk += 4) {
    const int ka = kk + khalf;
    v2f a, b;
    a.x = A[(size_t)m * K + ka];            // A: 16x4, row striped over 2 VGPRs
    a.y = A[(size_t)m * K + ka + 1];
    b.x = B[(size_t)ka * Nd + n];           // B: 4x16, K rows striped over 2 VGPRs
    b.y = B[(size_t)(ka + 1) * Nd + n];
    c = __builtin_amdgcn_wmma_f32_16x16x4_f32(false, a, false, b, (short)0, c,
                                              false, false);
  }

  const int rowBase = tileM * 16 + (lane >> 4) * 8;  // C/D layout: M = vgpr + 8*(lane>=16)
#pragma unroll
  for (int r = 0; r < 8; ++r) {
    C[(size_t)(rowBase + r) * Nd + n] = c[r];
  }
}

// ---------------------------------------------------------------------------
// Self-loop + optional bias init:  out[i*F+j] = (bias?bias[j]:0) + dinv[i]^2 * hw[i*F+j]
// vectorized float4; F % 4 == 0
// ---------------------------------------------------------------------------
__global__ void k_init_agg(const float* __restrict__ hw, const float* __restrict__ dinv,
                           const float* __restrict__ bias, float* __restrict__ out,
                           int n, int F) {
  const int fq  = F >> 2;
  long long idx = (long long)blockIdx.x * blockDim.x + threadIdx.x;
  if (idx >= (long long)n * fq) return;
  int i = (int)(idx / fq);
  int q = (int)(idx - (long long)i * fq);
  float di = dinv[i];
  float s  = di * di;  // self-loop norm: dinv[i] * 1.0 * dinv[i]
  float4 v = ((const float4*)hw)[idx];
  float4 b = bias ? ((const float4*)bias)[q] : make_float4(0.f, 0.f, 0.f, 0.f);
  float4 r = make_float4(fmaf(s, v.x, b.x), fmaf(s, v.y, b.y),
                         fmaf(s, v.z, b.z), fmaf(s, v.w, b.w));
  ((float4*)out)[idx] = r;
}

// ---------------------------------------------------------------------------
// Edge scatter-add, F=128: 32 lanes (one wave) per edge, float4 per lane.
// Index/norm loads are wave-uniform (broadcast); feature loads coalesced.
// ---------------------------------------------------------------------------
__global__ void k_agg_edges128(const int* __restrict__ src, const int* __restrict__ dst,
                               const float* __restrict__ norm, const float* __restrict__ hw,
                               float* __restrict__ out, int e) {
  long long idx = (long long)blockIdx.x * blockDim.x + threadIdx.x;
  int ei = (int)(idx >> 5);
  int q  = (int)(idx & 31);
  if (ei >= e) return;
  int s   = src[ei];
  int d   = dst[ei];
  float w = norm[ei];
  float4 v = *(const float4*)(hw + (size_t)s * 128 + q * 4);
  float* o = out + (size_t)d * 128 + q * 4;
  atomicAdd(o + 0, w * v.x);
  atomicAdd(o + 1, w * v.y);
  atomicAdd(o + 2, w * v.z);
  atomicAdd(o + 3, w * v.w);
}

// Fused mu/logstd edge scatter-add, F=64: 16 lanes per edge, both heads share
// the src/dst/norm loads.
__global__ void k_agg_edges64_pair(const int* __restrict__ src, const int* __restrict__ dst,
                                   const float* __restrict__ norm,
                                   const float* __restrict__ hmu, const float* __restrict__ hls,
                                   float* __restrict__ omu, float* __restrict__ ols, int e) {
  long long idx = (long long)blockIdx.x * blockDim.x + threadIdx.x;
  int ei = (int)(idx >> 4);
  int q  = (int)(idx & 15);
  if (ei >= e) return;
  int s   = src[ei];
  int d   = dst[ei];
  float w = norm[ei];
  float4 a = *(const float4*)(hmu + (size_t)s * 64 + q * 4);
  float4 b = *(const float4*)(hls + (size_t)s * 64 + q * 4);
  float* pm = omu + (size_t)d * 64 + q * 4;
  float* pl = ols + (size_t)d * 64 + q * 4;
  atomicAdd(pm + 0, w * a.x);
  atomicAdd(pm + 1, w * a.y);
  atomicAdd(pm + 2, w * a.z);
  atomicAdd(pm + 3, w * a.w);
  atomicAdd(pl + 0, w * b.x);
  atomicAdd(pl + 1, w * b.y);
  atomicAdd(pl + 2, w * b.z);
  atomicAdd(pl + 3, w * b.w);
}

// ELU(h + b1) in place, F must be a power of two (128)
__global__ void k_elu_bias(float* __restrict__ h, const float* __restrict__ bias,
                           long long total, int Fmask) {
  long long idx = (long long)blockIdx.x * blockDim.x + threadIdx.x;
  if (idx >= total) return;
  float v = h[idx] + bias[(int)(idx & Fmask)];
  h[idx] = v > 0.0f ? v : expm1f(v);
}

// ---------------------------------------------------------------------------
extern "C" void kernel_launch(void* const* d_in, const int* in_sizes, int n_in,
                              void* d_out, int out_size, void* d_ws, size_t ws_size,
                              hipStream_t stream) {
  const int IN_C = 256, HID = 128, OUT_C = 64;

  const float* x   = (const float*)d_in[0];
  const int*   eix = (const int*)d_in[1];
  const float* ew  = (const float*)d_in[2];
  const float* W1  = (const float*)d_in[3];
  const float* b1  = (const float*)d_in[4];
  const float* Wmu = (const float*)d_in[5];
  const float* bmu = (const float*)d_in[6];
  const float* Wls = (const float*)d_in[7];
  const float* bls = (const float*)d_in[8];

  const int N = in_sizes[0] / IN_C;  // 50000 (== 3125 * 16, no GEMM tail)
  const int E = in_sizes[2];         // 1600000

  const int* src = eix;
  const int* dst = eix + E;

  // ---- workspace layout (256B-aligned regions, xW region reused for hW) ----
  char*  ws  = (char*)d_ws;
  size_t off = 0;
  auto carve = [&](size_t bytes) -> char* {
    char* p = ws + off;
    off += (bytes + 255) & ~(size_t)255;
    return p;
  };
  float* dinv = (float*)carve((size_t)N * 4);            // deg -> dinv in place
  float* nrm  = (float*)carve((size_t)E * 4);            // per-edge norm
  size_t xwOff = off;
  float* xW   = (float*)carve((size_t)N * HID * 4);      // x @ W1
  float* h    = (float*)carve((size_t)N * HID * 4);      // agg1 -> ELU -> h
  float* hmu  = (float*)(ws + xwOff);                    // reuse xW region
  float* hls  = hmu + (size_t)N * OUT_C;

  float* out_mu = (float*)d_out;
  float* out_ls = out_mu + (size_t)N * OUT_C;

  const int B = 256;
  auto blocks = [](long long t, int b) { return (unsigned)((t + b - 1) / b); };

  // 1) symmetric GCN normalization
  k_init_deg<<<blocks(N, B), B, 0, stream>>>(dinv, N);
  k_deg_edges<<<blocks(E, B), B, 0, stream>>>(dst, ew, dinv, E);
  k_dinv<<<blocks(N, B), B, 0, stream>>>(dinv, N);
  k_edge_norm<<<blocks(E, B), B, 0, stream>>>(src, dst, ew, dinv, nrm, E);

  // 2) layer 1: xW = x @ W1 (WMMA), then self-loop init + edge scatter + ELU
  k_gemm_f32_wmma<<<dim3(N / 16, HID / 64), dim3(128), 0, stream>>>(x, W1, xW, N, IN_C, HID);
  k_init_agg<<<blocks((long long)N * (HID / 4), B), B, 0, stream>>>(xW, dinv, nullptr, h, N, HID);
  k_agg_edges128<<<blocks((long long)E * 32, B), B, 0, stream>>>(src, dst, nrm, xW, h, E);
  k_elu_bias<<<blocks((long long)N * HID, B), B, 0, stream>>>(h, b1, (long long)N * HID, HID - 1);

  // 3) layer 2 heads: hW = h @ W (WMMA), init d_out with bias + self-loop,
  //    then fused mu/logstd edge scatter directly into d_out
  k_gemm_f32_wmma<<<dim3(N / 16, 1), dim3(128), 0, stream>>>(h, Wmu, hmu, N, HID, OUT_C);
  k_gemm_f32_wmma<<<dim3(N / 16, 1), dim3(128), 0, stream>>>(h, Wls, hls, N, HID, OUT_C);
  k_init_agg<<<blocks((long long)N * (OUT_C / 4), B), B, 0, stream>>>(hmu, dinv, bmu, out_mu, N, OUT_C);
  k_init_agg<<<blocks((long long)N * (OUT_C / 4), B), B, 0, stream>>>(hls, dinv, bls, out_ls, N, OUT_C);
  k_agg_edges64_pair<<<blocks((long long)E * 16, B), B, 0, stream>>>(src, dst, nrm, hmu, hls,
                                                                     out_mu, out_ls, E);
}